// R_HGNN_Layer_48009144434782
// MI455X (gfx1250) — compile-verified
//
#include <hip/hip_runtime.h>
#include <math.h>

// ---------------------------------------------------------------------------
// R-HGNN layer for MI455X (gfx1250, wave32).
// GEMMs ([50000,64]@[64,256]) use V_WMMA_F32_16X16X4_F32 (f32 precision, K=4
// per step, 16 steps for K=64).  Edge attention / message passing is the
// bandwidth-bound part (fits in 192MB L2) and uses plain VALU + atomics.
// ---------------------------------------------------------------------------

#define NN      50000     // N_A == N_B
#define EE      250000    // edges per relation
#define INDIM   64
#define NH      8         // heads
#define ND      32        // dim per head
#define HD      256       // NH*ND
#define SLOPE   0.2f

typedef float v2f __attribute__((ext_vector_type(2)));
typedef float v8f __attribute__((ext_vector_type(8)));

// ---------------------------------------------------------------------------
// C[M,256] = A[M,64] @ B[64,256]   (row-major, f32, WMMA 16x16x4)
// block = 256 threads = 8 waves; each wave owns one 16x16 C tile.
// grid  = (M/16, 2) -> blockIdx.y*8+wave selects one of the 16 column tiles.
// Per-lane layouts follow CDNA5 ISA 7.12.2:
//   A 16x4 f32 : lanes 0-15 M=0..15 {K=k0,k0+1}, lanes 16-31 M=0..15 {K=k0+2,k0+3}
//   B 4x16 f32 : lanes 0-15 rows k0,k0+1 (N=n0+lane), lanes 16-31 rows k0+2,k0+3
//   C/D 16x16  : VGPR r -> lanes 0-15 M=r, lanes 16-31 M=r+8, N = lane%16
// ---------------------------------------------------------------------------
__global__ __launch_bounds__(256)
void gemm64x256_wmma(const float* __restrict__ A, const float* __restrict__ B,
                     float* __restrict__ C, int M) {
  const int wave = threadIdx.x >> 5;
  const int lane = threadIdx.x & 31;
  const int half = lane >> 4;      // 0: lanes 0-15, 1: lanes 16-31
  const int lm   = lane & 15;
  const int m0   = blockIdx.x << 4;
  const int n0   = (((int)blockIdx.y << 3) + wave) << 4;
  if (m0 + 15 >= M + 16) return;   // M is a multiple of 16 here (50000 = 3125*16)

  const float* arow = A + (size_t)(m0 + lm) * INDIM;
  v8f acc = {};
#pragma unroll
  for (int k0 = 0; k0 < INDIM; k0 += 4) {
    const int ka = k0 + (half << 1);
    v2f a, b;
    a.x = arow[ka];
    a.y = arow[ka + 1];
    b.x = B[(size_t)(ka)     * HD + n0 + lm];
    b.y = B[(size_t)(ka + 1) * HD + n0 + lm];
    acc = __builtin_amdgcn_wmma_f32_16x16x4_f32(false, a, false, b,
                                                (short)0, acc, false, false);
  }
  float* crow = C + (size_t)m0 * HD + n0 + lm;
#pragma unroll
  for (int r = 0; r < 8; ++r)
    crow[(size_t)(r + (half << 3)) * HD] = acc[r];
}

// ---------------------------------------------------------------------------
// raw[512] = rel_emb[64] @ W_rel[64,512];  rel_out[256] = rel_emb @ W_prop + b
// one block of 512 threads per relation
// ---------------------------------------------------------------------------
__global__ void rel_vec_kernel(const float* __restrict__ rel_emb,
                               const float* __restrict__ W_rel,
                               const float* __restrict__ W_prop,
                               const float* __restrict__ b_prop,
                               float* __restrict__ raw,
                               float* __restrict__ rel_out) {
  const int j = threadIdx.x;
  if (j < 512) {
    float s = 0.f;
#pragma unroll 8
    for (int k = 0; k < 64; ++k) s += rel_emb[k] * W_rel[k * 512 + j];
    raw[j] = s;
  }
  if (j < 256) {
    float s = 0.f;
#pragma unroll 8
    for (int k = 0; k < 64; ++k) s += rel_emb[k] * W_prop[k * 256 + j];
    rel_out[j] = s + b_prop[j];
  }
}

// per-(node,head) attention logits:  e_src = <fs[n,h,:], raw[h,32:]>, e_dst = <fd, raw[h,:32]>
__global__ void node_scores_kernel(const float* __restrict__ fs,
                                   const float* __restrict__ fd,
                                   const float* __restrict__ raw,
                                   float* __restrict__ esrc,
                                   float* __restrict__ edst, int n) {
  const int t = blockIdx.x * blockDim.x + threadIdx.x;
  if (t >= n * NH) return;
  const int node = t >> 3, h = t & 7;
  const float* rd = raw + h * 64;       // first D  -> dst
  const float* rs = rd + ND;            // second D -> src
  const float* ps = fs + (size_t)node * HD + h * ND;
  const float* pd = fd + (size_t)node * HD + h * ND;
  float ss = 0.f, sd = 0.f;
#pragma unroll
  for (int d = 0; d < ND; ++d) { ss += ps[d] * rs[d]; sd += pd[d] * rd[d]; }
  esrc[t] = ss;
  edst[t] = sd;
}

__global__ void fill_f32_kernel(float* p, float v, size_t n) {
  size_t i = (size_t)blockIdx.x * blockDim.x + threadIdx.x;
  if (i < n) p[i] = v;
}
__global__ void fill_u32_kernel(unsigned int* p, unsigned int v, size_t n) {
  size_t i = (size_t)blockIdx.x * blockDim.x + threadIdx.x;
  if (i < n) p[i] = v;
}

// monotonic float->uint encoding so plain integer atomicMax does float max
__device__ __forceinline__ unsigned int enc_f32(float f) {
  unsigned int u = __float_as_uint(f);
  return (u & 0x80000000u) ? ~u : (u | 0x80000000u);
}
__device__ __forceinline__ float dec_f32(unsigned int u) {
  return __uint_as_float((u & 0x80000000u) ? (u & 0x7FFFFFFFu) : ~u);
}
#define ENC_NEG_INF 0x007FFFFFu   // enc(-inf)

// pass 1: e = leaky_relu(esrc[src]+edst[dst]); store; segment-max into emax[dst]
__global__ void edge_logits_kernel(const int* __restrict__ src,
                                   const int* __restrict__ dst,
                                   const float* __restrict__ esrc,
                                   const float* __restrict__ edst,
                                   float* __restrict__ ew,
                                   unsigned int* __restrict__ emax, int ne) {
  const int t = blockIdx.x * blockDim.x + threadIdx.x;
  if (t >= ne * NH) return;
  const int i = t >> 3, h = t & 7;
  float e = esrc[src[i] * NH + h] + edst[dst[i] * NH + h];
  e = (e > 0.f) ? e : SLOPE * e;
  ew[t] = e;
  atomicMax(&emax[dst[i] * NH + h], enc_f32(e));
}

// pass 2: w = exp(e - emax[dst]); store; segment-sum into denom[dst]
__global__ void edge_exp_kernel(const int* __restrict__ dst,
                                float* __restrict__ ew,
                                const unsigned int* __restrict__ emax,
                                float* __restrict__ denom, int ne) {
  const int t = blockIdx.x * blockDim.x + threadIdx.x;
  if (t >= ne * NH) return;
  const int i = t >> 3, h = t & 7;
  const float m = dec_f32(emax[dst[i] * NH + h]);
  const float w = expf(ew[t] - m);
  ew[t] = w;
  atomicAdd(&denom[dst[i] * NH + h], w);
}

// pass 3: acc[dst, c] += fs[src, c] * (w / max(denom,1e-9)); one thread per (edge, channel)
__global__ void edge_message_kernel(const int* __restrict__ src,
                                    const int* __restrict__ dst,
                                    const float* __restrict__ fs,
                                    const float* __restrict__ ew,
                                    const float* __restrict__ denom,
                                    float* __restrict__ acc, int ne) {
  const long long t = (long long)blockIdx.x * blockDim.x + threadIdx.x;
  if (t >= (long long)ne * HD) return;
  const int i = (int)(t >> 8);
  const int c = (int)(t & 255);
  const int h = c >> 5;
  const int d_ = dst[i];
  const float a = ew[i * NH + h] / fmaxf(denom[d_ * NH + h], 1e-9f);
  atomicAdd(&acc[(size_t)d_ * HD + c], fs[(size_t)src[i] * HD + c] * a);
}

// fuse: relu -> gated residual -> relation-crossing softmax over the 2
// relations sharing this dst type, using this output relation's cross_w.
__global__ void fuse_kernel(const float* __restrict__ acc0, const float* __restrict__ res0,
                            const float* __restrict__ acc1, const float* __restrict__ res1,
                            const float* __restrict__ b_res, const float* __restrict__ res_w,
                            const float* __restrict__ cross_w,
                            float* __restrict__ out, int n) {
  const int t = blockIdx.x * blockDim.x + threadIdx.x;
  if (t >= n * NH) return;
  const int node = t >> 3, h = t & 7;
  const float alpha = 1.f / (1.f + expf(-res_w[0]));
  const float beta  = 1.f - alpha;
  const size_t base = (size_t)node * HD + h * ND;
  const float* cw = cross_w + h * ND;
  const float* br = b_res + h * ND;
  float x0[ND], x1[ND];
  float s0 = 0.f, s1 = 0.f;
#pragma unroll
  for (int d = 0; d < ND; ++d) {
    const float o0 = fmaxf(acc0[base + d], 0.f);
    const float o1 = fmaxf(acc1[base + d], 0.f);
    const float v0 = alpha * o0 + beta * (res0[base + d] + br[d]);
    const float v1 = alpha * o1 + beta * (res1[base + d] + br[d]);
    x0[d] = v0; x1[d] = v1;
    s0 += v0 * cw[d]; s1 += v1 * cw[d];
  }
  s0 = (s0 > 0.f) ? s0 : SLOPE * s0;
  s1 = (s1 > 0.f) ? s1 : SLOPE * s1;
  const float m  = fmaxf(s0, s1);
  float w0 = expf(s0 - m), w1 = expf(s1 - m);
  const float inv = 1.f / (w0 + w1);
  w0 *= inv; w1 *= inv;
#pragma unroll
  for (int d = 0; d < ND; ++d)
    out[base + d] = x0[d] * w0 + x1[d] * w1;
}

// ---------------------------------------------------------------------------
// Host-side launch.
//
// Input pointer map (setup_inputs order; params flattened as a jax pytree,
// i.e. dict keys in sorted order: W_node, W_prop, W_rel, W_res, b_prop,
// b_res, cross_w, rel_emb, res_w; inner dicts sorted = insertion order):
//  0 feat_B_ab1  1 feat_B_ab2  2 feat_A_ba1  3 feat_A_ba2
//  4 src_ab1 5 dst_ab1 6 src_ab2 7 dst_ab2 8 src_ba1 9 dst_ba1 10 src_ba2 11 dst_ba2
// 12 W_node_A 13 W_node_B
// 14..17 W_prop_{ab1,ab2,ba1,ba2}   18..21 W_rel_{...}
// 22 W_res_A 23 W_res_B             24..27 b_prop_{...}
// 28 b_res_A 29 b_res_B             30..33 cross_w_{...}
// 34..37 rel_emb_{...}              38 res_w_A 39 res_w_B
// ---------------------------------------------------------------------------
extern "C" void kernel_launch(void* const* d_in, const int* in_sizes, int n_in,
                              void* d_out, int out_size, void* d_ws, size_t ws_size,
                              hipStream_t stream) {
  const float* featB1 = (const float*)d_in[0];
  const float* featB2 = (const float*)d_in[1];
  const float* featA1 = (const float*)d_in[2];
  const float* featA2 = (const float*)d_in[3];
  const int* src[4] = {(const int*)d_in[4], (const int*)d_in[6],
                       (const int*)d_in[8], (const int*)d_in[10]};
  const int* dst[4] = {(const int*)d_in[5], (const int*)d_in[7],
                       (const int*)d_in[9], (const int*)d_in[11]};
  const float* W_node_A = (const float*)d_in[12];
  const float* W_node_B = (const float*)d_in[13];
  const float* W_prop[4] = {(const float*)d_in[14], (const float*)d_in[15],
                            (const float*)d_in[16], (const float*)d_in[17]};
  const float* W_rel[4]  = {(const float*)d_in[18], (const float*)d_in[19],
                            (const float*)d_in[20], (const float*)d_in[21]};
  const float* W_res_A = (const float*)d_in[22];
  const float* W_res_B = (const float*)d_in[23];
  const float* b_prop[4] = {(const float*)d_in[24], (const float*)d_in[25],
                            (const float*)d_in[26], (const float*)d_in[27]};
  const float* b_res_A = (const float*)d_in[28];
  const float* b_res_B = (const float*)d_in[29];
  const float* cross_w[4] = {(const float*)d_in[30], (const float*)d_in[31],
                             (const float*)d_in[32], (const float*)d_in[33]};
  const float* rel_emb[4] = {(const float*)d_in[34], (const float*)d_in[35],
                             (const float*)d_in[36], (const float*)d_in[37]};
  const float* res_w_A = (const float*)d_in[38];
  const float* res_w_B = (const float*)d_in[39];

  float* out = (float*)d_out;

  // --- workspace carve-up ---------------------------------------------------
  const size_t NHD = (size_t)NN * HD;   // 12.8M floats
  const size_t NHh = (size_t)NN * NH;   // 400K floats
  const size_t EHh = (size_t)EE * NH;   // 2M floats
  char* w = (char*)d_ws;
  float* P   = (float*)w;  w += 4 * NHD * sizeof(float);   // PA1,PA2,PB1,PB2
  float* R   = (float*)w;  w += 4 * NHD * sizeof(float);   // res ab1,ab2,ba1,ba2
  float* ACC = (float*)w;  w += 4 * NHD * sizeof(float);   // message accumulators
  float* RAW = (float*)w;  w += 4 * 512 * sizeof(float);
  float* ESRC= (float*)w;  w += 4 * NHh * sizeof(float);
  float* EDST= (float*)w;  w += 4 * NHh * sizeof(float);
  unsigned int* EMAX = (unsigned int*)w; w += 4 * NHh * sizeof(unsigned int);
  float* DEN = (float*)w;  w += 4 * NHh * sizeof(float);
  float* EW  = (float*)w;  w += 4 * EHh * sizeof(float);
  (void)ws_size; (void)n_in; (void)in_sizes; (void)out_size;

  float* P0 = P;            // feat_A_ba1 @ W_node_A  (fs ab1 / fd ba1)
  float* P1 = P + NHD;      // feat_A_ba2 @ W_node_A
  float* P2 = P + 2 * NHD;  // feat_B_ab1 @ W_node_B
  float* P3 = P + 3 * NHD;  // feat_B_ab2 @ W_node_B

  const dim3 gGemm(NN / 16, 2), bGemm(256);
  // node projections (4 distinct; fs of e == fd of REV[e])
  gemm64x256_wmma<<<gGemm, bGemm, 0, stream>>>(featA1, W_node_A, P0, NN);
  gemm64x256_wmma<<<gGemm, bGemm, 0, stream>>>(featA2, W_node_A, P1, NN);
  gemm64x256_wmma<<<gGemm, bGemm, 0, stream>>>(featB1, W_node_B, P2, NN);
  gemm64x256_wmma<<<gGemm, bGemm, 0, stream>>>(featB2, W_node_B, P3, NN);
  // residual projections (bias added in fuse)
  gemm64x256_wmma<<<gGemm, bGemm, 0, stream>>>(featB1, W_res_B, R + 0 * NHD, NN);
  gemm64x256_wmma<<<gGemm, bGemm, 0, stream>>>(featB2, W_res_B, R + 1 * NHD, NN);
  gemm64x256_wmma<<<gGemm, bGemm, 0, stream>>>(featA1, W_res_A, R + 2 * NHD, NN);
  gemm64x256_wmma<<<gGemm, bGemm, 0, stream>>>(featA2, W_res_A, R + 3 * NHD, NN);

  // relation vectors: raw (attention weights) + rel_out tail of d_out
  float* rel_out = out + 4 * NHD;
  for (int e = 0; e < 4; ++e)
    rel_vec_kernel<<<1, 512, 0, stream>>>(rel_emb[e], W_rel[e], W_prop[e],
                                          b_prop[e], RAW + e * 512,
                                          rel_out + e * HD);

  // per-node attention logits: fs/fd per relation
  const float* FS[4] = {P0, P1, P2, P3};
  const float* FD[4] = {P2, P3, P0, P1};
  const int nhBlocks = (int)((NHh + 255) / 256);
  for (int e = 0; e < 4; ++e)
    node_scores_kernel<<<nhBlocks, 256, 0, stream>>>(FS[e], FD[e], RAW + e * 512,
                                                     ESRC + e * NHh, EDST + e * NHh, NN);

  // init accumulators (d_ws is poisoned by the harness -> must clear each call)
  fill_f32_kernel<<<(int)((4 * NHD + 255) / 256), 256, 0, stream>>>(ACC, 0.f, 4 * NHD);
  fill_f32_kernel<<<(int)((4 * NHh + 255) / 256), 256, 0, stream>>>(DEN, 0.f, 4 * NHh);
  fill_u32_kernel<<<(int)((4 * NHh + 255) / 256), 256, 0, stream>>>(EMAX, ENC_NEG_INF, 4 * NHh);

  const int ehBlocks = (int)((EHh + 255) / 256);
  for (int e = 0; e < 4; ++e)
    edge_logits_kernel<<<ehBlocks, 256, 0, stream>>>(src[e], dst[e],
                                                     ESRC + e * NHh, EDST + e * NHh,
                                                     EW + e * EHh, EMAX + e * NHh, EE);
  for (int e = 0; e < 4; ++e)
    edge_exp_kernel<<<ehBlocks, 256, 0, stream>>>(dst[e], EW + e * EHh,
                                                  EMAX + e * NHh, DEN + e * NHh, EE);
  const long long msgThreads = (long long)EE * HD;
  const int msgBlocks = (int)((msgThreads + 255) / 256);
  for (int e = 0; e < 4; ++e)
    edge_message_kernel<<<msgBlocks, 256, 0, stream>>>(src[e], dst[e], FS[e],
                                                       EW + e * EHh, DEN + e * NHh,
                                                       ACC + e * NHD, EE);

  // relation crossing + gated residual -> final node_out (RELS order)
  // dst B pair: (ab1, ab2) = ACC[0],ACC[1]; dst A pair: (ba1, ba2) = ACC[2],ACC[3]
  fuse_kernel<<<nhBlocks, 256, 0, stream>>>(ACC + 0 * NHD, R + 0 * NHD,
                                            ACC + 1 * NHD, R + 1 * NHD,
                                            b_res_B, res_w_B, cross_w[0],
                                            out + 0 * NHD, NN);
  fuse_kernel<<<nhBlocks, 256, 0, stream>>>(ACC + 0 * NHD, R + 0 * NHD,
                                            ACC + 1 * NHD, R + 1 * NHD,
                                            b_res_B, res_w_B, cross_w[1],
                                            out + 1 * NHD, NN);
  fuse_kernel<<<nhBlocks, 256, 0, stream>>>(ACC + 2 * NHD, R + 2 * NHD,
                                            ACC + 3 * NHD, R + 3 * NHD,
                                            b_res_A, res_w_A, cross_w[2],
                                            out + 2 * NHD, NN);
  fuse_kernel<<<nhBlocks, 256, 0, stream>>>(ACC + 2 * NHD, R + 2 * NHD,
                                            ACC + 3 * NHD, R + 3 * NHD,
                                            b_res_A, res_w_A, cross_w[3],
                                            out + 3 * NHD, NN);
}